// AdaptivePooling2D_88768384074199
// MI455X (gfx1250) — compile-verified
//
#include <hip/hip_runtime.h>
#include <math.h>

typedef __attribute__((ext_vector_type(2))) float v2f;
typedef __attribute__((ext_vector_type(4))) float v4f;
typedef __attribute__((ext_vector_type(8))) float v8f;

#define B_  16
#define H_  225
#define W_  225
#define C_  256
#define OX_ 7
#define OY_ 7

// Reproduce TF/reference fp32 index math: start = int32(o * (in/out)),
// end = int32(ceil((o+1) * (in/out))) with scale computed in fp32.
__device__ __forceinline__ int win_start(int o) {
    const float scale = 225.0f / 7.0f;
    return (int)((float)o * scale);
}
__device__ __forceinline__ int win_end(int o) {
    const float scale = 225.0f / 7.0f;
    return (int)ceilf(((float)o + 1.0f) * scale);
}

// ---------------------------------------------------------------------------
// Stage 1: pool over H.  mid[b, ox, w, c] = mean_{h in win(ox)} in[b, h, w, c]
// Streaming stage (~95% of total traffic). Non-temporal b128 loads: the
// input is touched exactly once, so keep it out of L2 — the 24.6 MB mid
// buffer then stays L2-resident for stage 2 (192 MB L2).
// grid = (ceil(W/4), OX, B), block = 256
// ---------------------------------------------------------------------------
__global__ void pool_rows_kernel(const float* __restrict__ in,
                                 float* __restrict__ mid) {
    const int t   = threadIdx.x;
    const int sub = t >> 6;        // 0..3: which w within the group of 4
    const int c4  = t & 63;        // float4 index within the 256 channels
    const int w   = blockIdx.x * 4 + sub;
    const int ox  = blockIdx.y;
    const int b   = blockIdx.z;
    if (w >= W_) return;

    const int   hs  = win_start(ox);
    const int   he  = win_end(ox);
    const float inv = 1.0f / (float)(he - hs);

    const v4f* in4 = (const v4f*)in;
    const long hstride = (long)W_ * (C_ / 4);
    long idx = ((long)(b * H_ + hs) * W_ + w) * (C_ / 4) + c4;

    v4f acc = (v4f)(0.0f);
#pragma unroll 4
    for (int h = hs; h < he; ++h) {
        v4f v = __builtin_nontemporal_load(&in4[idx]);
        acc += v;
        idx += hstride;
    }
    acc *= inv;

    v4f* mid4 = (v4f*)mid;
    mid4[((long)(b * OX_ + ox) * W_ + w) * (C_ / 4) + c4] = acc;
}

// ---------------------------------------------------------------------------
// Stage 2: pool over W as a fp32 WMMA matmul per (b, ox):
//   out[oy, c] = sum_w Py[oy, w] * mid[b, ox, w, c]
// M = 16 (7 rows live), K = 228 (225 live, 57 steps of 4), N = 256 (16 tiles).
// One wave per 16-column N-tile; 56 unguarded k-steps + 1 peeled tail step.
// A-fragment weights are generated branch-free with a single unsigned range
// compare per component (wlen == 0 encodes the zero padding rows m >= 7).
//
// Fragment layouts (ISA 7.12.2, 32-bit, wave32):
//   A 16x4 : lane m = lane&15; VGPR0 holds K = k0 + (lane<16 ? 0 : 2),
//            VGPR1 holds K+1.
//   B 4x16 : lane col n = lane&15; VGPR0 holds K = k0 + (lane<16 ? 0 : 2),
//            VGPR1 holds K+1.
//   D 16x16: VGPR j -> row M=j (lanes 0-15), M=j+8 (lanes 16-31), col = lane&15.
// grid = (B*OX*2), block = 256 (8 waves; 2 blocks cover the 16 N-tiles)
// ---------------------------------------------------------------------------
__global__ void pool_cols_wmma_kernel(const float* __restrict__ mid,
                                      float* __restrict__ out) {
    const int lane  = threadIdx.x & 31;
    const int wave  = threadIdx.x >> 5;                 // 0..7
    const int ntile = ((blockIdx.x & 1) << 3) + wave;   // 0..15
    const int bx    = blockIdx.x >> 1;                  // 0..B*OX-1
    const int b     = bx / OX_;
    const int ox    = bx % OX_;

    const int m     = lane & 15;
    const int khalf = (lane >> 4) << 1;                 // 0 or 2
    const int n     = (ntile << 4) + m;                 // B/D column of this lane

    // Per-lane window for A. wlen = 0 for padding rows (m >= 7): the unsigned
    // range test (k - ws) < wlen is then always false -> weight 0.
    int ws = 0;
    unsigned wlen = 0;
    float invw = 0.0f;
    if (m < OY_) {
        ws   = win_start(m);
        wlen = (unsigned)(win_end(m) - ws);
        invw = 1.0f / (float)wlen;
    }

    const float* src = mid + (long)(b * OX_ + ox) * W_ * C_ + n;

    v8f acc = (v8f)(0.0f);
    // Main loop: ks = 0..55 -> max k touched is 222+1 = 223 < 225: no guards.
    for (int ks = 0; ks < 56; ++ks) {
        const int kx = (ks << 2) + khalf;
        const int ky = kx + 1;

        v2f a, bm;
        a.x  = ((unsigned)(kx - ws) < wlen) ? invw : 0.0f;
        a.y  = ((unsigned)(ky - ws) < wlen) ? invw : 0.0f;
        bm.x = src[(long)kx * C_];
        bm.y = src[(long)ky * C_];

        acc = __builtin_amdgcn_wmma_f32_16x16x4_f32(
            /*neg_a=*/false, a, /*neg_b=*/false, bm,
            /*c_mod=*/(short)0, acc, /*reuse_a=*/false, /*reuse_b=*/false);
    }
    // Peeled tail: ks = 56 covers k = 224..227; only k = 224 (lo half, .x) is
    // live. A-weights fall out of the same unsigned range test automatically.
    {
        const int kx = 224 + khalf;
        v2f a, bm;
        a.x  = ((unsigned)(kx - ws) < wlen) ? invw : 0.0f;
        a.y  = ((unsigned)(kx + 1 - ws) < wlen) ? invw : 0.0f;
        bm.x = (khalf == 0) ? src[(long)224 * C_] : 0.0f;
        bm.y = 0.0f;

        acc = __builtin_amdgcn_wmma_f32_16x16x4_f32(
            false, a, false, bm, (short)0, acc, false, false);
    }

    // Rows 0..6 of D live in VGPRs 0..6 on lanes 0-15.
    if (lane < 16) {
        float* dst = out + (long)(b * OX_ + ox) * OY_ * C_ + (ntile << 4) + lane;
#pragma unroll
        for (int j = 0; j < OY_; ++j) dst[(long)j * C_] = acc[j];
    }
}

// ---------------------------------------------------------------------------
// Fallback (only if the workspace is too small): fused direct pooling.
// grid = (OX*OY, B), block = 256 (one channel per thread)
// ---------------------------------------------------------------------------
__global__ void pool_direct_kernel(const float* __restrict__ in,
                                   float* __restrict__ out) {
    const int c  = threadIdx.x;
    const int oy = blockIdx.x % OY_;
    const int ox = blockIdx.x / OY_;
    const int b  = blockIdx.y;

    const int hs = win_start(ox), he = win_end(ox);
    const int wss = win_start(oy), wse = win_end(oy);
    const float inv = 1.0f / ((float)(he - hs) * (float)(wse - wss));

    float s = 0.0f;
    for (int h = hs; h < he; ++h) {
        const float* row = in + ((long)(b * H_ + h) * W_) * C_ + c;
        for (int w = wss; w < wse; ++w) s += row[(long)w * C_];
    }
    out[((long)(b * OX_ + ox) * OY_ + oy) * C_ + c] = s * inv;
}

extern "C" void kernel_launch(void* const* d_in, const int* in_sizes, int n_in,
                              void* d_out, int out_size, void* d_ws, size_t ws_size,
                              hipStream_t stream) {
    (void)in_sizes; (void)n_in; (void)out_size;
    const float* in  = (const float*)d_in[0];
    float*       out = (float*)d_out;

    const size_t mid_bytes = (size_t)B_ * OX_ * W_ * C_ * sizeof(float); // ~24.6 MB
    if (ws_size >= mid_bytes) {
        float* mid = (float*)d_ws;
        dim3 g1((W_ + 3) / 4, OX_, B_);
        pool_rows_kernel<<<g1, 256, 0, stream>>>(in, mid);
        pool_cols_wmma_kernel<<<dim3(B_ * OX_ * 2), 256, 0, stream>>>(mid, out);
    } else {
        pool_direct_kernel<<<dim3(OX_ * OY_, B_), C_, 0, stream>>>(in, out);
    }
}